// YOLOv3_2911987827134
// MI455X (gfx1250) — compile-verified
//
#include <hip/hip_runtime.h>
#include <hip/hip_bf16.h>
#include <stdint.h>

typedef __attribute__((ext_vector_type(2))) float v2f;
typedef __attribute__((ext_vector_type(8))) float v8f;
typedef __attribute__((ext_vector_type(4))) unsigned int v4u;
typedef __attribute__((ext_vector_type(4))) int v4i;
typedef __attribute__((ext_vector_type(8))) int v8i;

#define B_   32
#define CH_  28
#define HW3_ 9216
#define HW2_ 2304
#define HW1_ 576
#define A3_  36864      // 4*9216
#define A2_  9216       // 4*2304
#define A1_  2304       // 4*576
#define A_   48384
#define K_   1024
#define BG_BITS 0x3F000000u   // bits of 0.5f

#if defined(__has_builtin)
#if __has_builtin(__builtin_amdgcn_tensor_load_to_lds) && __has_builtin(__builtin_amdgcn_s_wait_tensorcnt)
#define HAVE_TDM 1
#endif
#endif

// ---- map global anchor index -> base pointer of its 7 channels (stride hw) ----
__device__ __forceinline__ const float* chan_base(const float* __restrict__ f3,
                                                  const float* __restrict__ f2,
                                                  const float* __restrict__ f1,
                                                  int b, unsigned a, int* hw_out) {
  if (a < A3_) {
    int slot = a / HW3_, pix = a - slot * HW3_;
    *hw_out = HW3_;
    return f3 + ((size_t)b * CH_ + slot * 7) * (size_t)HW3_ + pix;
  } else if (a < A3_ + A2_) {
    unsigned l = a - A3_;
    int slot = l / HW2_, pix = l - slot * HW2_;
    *hw_out = HW2_;
    return f2 + ((size_t)b * CH_ + slot * 7) * (size_t)HW2_ + pix;
  } else {
    unsigned l = a - (A3_ + A2_);
    int slot = l / HW1_, pix = l - slot * HW1_;
    *hw_out = HW1_;
    return f1 + ((size_t)b * CH_ + slot * 7) * (size_t)HW1_ + pix;
  }
}

__device__ __forceinline__ unsigned ballot32(bool p) {
  return __builtin_amdgcn_ballot_w32(p);
}

// =============== Kernel 1: objectness score keys =================
__global__ void k_scores(const float* __restrict__ f3, const float* __restrict__ f2,
                         const float* __restrict__ f1, unsigned* __restrict__ keys) {
  int t = blockIdx.x * blockDim.x + threadIdx.x;
  if (t >= B_ * A_) return;
  int b = t / A_;
  unsigned a = (unsigned)(t - b * A_);
  int hw;
  const float* ch = chan_base(f3, f2, f1, b, a, &hw);
  __builtin_prefetch(ch, 0, 3);              // global_prefetch_b8 (decode reads this later)
  float d4 = ch[4 * (size_t)hw];
  float s = fminf(fmaxf(d4, 0.0f), 1.0f);
  // score = obj if obj>=0.5 else -1. Key: positive-float bits are order-preserving.
  keys[t] = (s >= 0.5f) ? __float_as_uint(s) : 0u;
}

// =============== Kernel 2: exact top-K (value desc, index asc) per batch =========
__global__ __launch_bounds__(1024) void k_select(const unsigned* __restrict__ keys,
                                                 unsigned* __restrict__ topidx,
                                                 unsigned* __restrict__ validf) {
  __shared__ unsigned hist[256];
  __shared__ unsigned sh_prefix, sh_need;
  __shared__ unsigned wtot[64];            // [0..31] gt totals, [32..63] eq totals
  __shared__ unsigned sh_base[2];
  __shared__ unsigned long long sel[K_];   // stores ~key64 (ascending sort => key64 desc)

  int b = blockIdx.x;
  const unsigned* kb = keys + (size_t)b * A_;
  int t = threadIdx.x, lane = t & 31, wid = t >> 5;

  // ---- Phase A: 4-level radix select of K-th largest key ----
  unsigned prefix = 0, need = K_;
  for (int l = 0; l < 4; ++l) {
    int shift = 24 - 8 * l;
    if (t < 256) hist[t] = 0;
    __syncthreads();
    unsigned pmask = (l == 0) ? 0u : (0xFFFFFFFFu << (shift + 8));
    for (int i = t; i < A_; i += 1024) {
      unsigned key = kb[i];
      if ((key & pmask) == prefix) atomicAdd(&hist[(key >> shift) & 255u], 1u);
    }
    __syncthreads();
    if (t == 0) {
      unsigned cum = 0;
      int bin = 255;
      for (; bin >= 0; --bin) {
        unsigned h = hist[bin];
        if (cum + h >= need) break;
        cum += h;
      }
      if (bin < 0) bin = 0;
      sh_prefix = prefix | ((unsigned)bin << shift);
      sh_need = need - cum;
    }
    __syncthreads();
    prefix = sh_prefix;
    need = sh_need;
    __syncthreads();
  }
  unsigned pivot = prefix;
  unsigned need_eq = need;              // # of ==pivot entries to take, lowest index first
  unsigned cnt_gt = K_ - need_eq;

  // ---- Phase B: ordered compaction (index-ascending chunks) ----
  if (t == 0) { sh_base[0] = 0; sh_base[1] = 0; }
  __syncthreads();
  const unsigned ltmask = (1u << lane) - 1u;
  for (int c = 0; c < (A_ + 1023) / 1024; ++c) {
    int i = c * 1024 + t;
    unsigned key = (i < A_) ? kb[i] : 0u;
    bool gt = (i < A_) && (key > pivot);
    bool eq = (i < A_) && (key == pivot);
    unsigned mg = ballot32(gt);
    unsigned me = ballot32(eq);
    unsigned rg = __popc(mg & ltmask);
    unsigned re = __popc(me & ltmask);
    if (lane == 0) { wtot[wid] = __popc(mg); wtot[32 + wid] = __popc(me); }
    __syncthreads();
    unsigned wbg = 0, wbe = 0;
    for (int w = 0; w < wid; ++w) { wbg += wtot[w]; wbe += wtot[32 + w]; }
    unsigned bg = sh_base[0], be = sh_base[1];
    unsigned long long key64 = ((unsigned long long)key << 32) | (unsigned)(~(unsigned)i);
    if (gt) sel[bg + wbg + rg] = ~key64;
    if (eq) {
      unsigned r = be + wbe + re;
      if (r < need_eq) sel[cnt_gt + r] = ~key64;
    }
    __syncthreads();
    if (t == 0) {
      unsigned tg = 0, te = 0;
      for (int w = 0; w < 32; ++w) { tg += wtot[w]; te += wtot[32 + w]; }
      sh_base[0] = bg + tg;
      sh_base[1] = be + te;
    }
    __syncthreads();
  }

  // ---- Phase C: bitonic ascending sort of ~key64 == key desc, index asc ----
  for (int kk = 2; kk <= K_; kk <<= 1) {
    for (int j = kk >> 1; j > 0; j >>= 1) {
      __syncthreads();
      int ixj = t ^ j;
      if (ixj > t) {
        unsigned long long a0 = sel[t], a1 = sel[ixj];
        bool up = ((t & kk) == 0);
        if ((a0 > a1) == up) { sel[t] = a1; sel[ixj] = a0; }
      }
    }
  }
  __syncthreads();

  unsigned long long e = ~sel[t];
  unsigned key = (unsigned)(e >> 32);
  unsigned idx = ~(unsigned)e;
  topidx[(size_t)b * K_ + t] = idx;
  validf[(size_t)b * K_ + t] = (key >= BG_BITS) ? 1u : 0u;
}

// =============== Kernel 3: gather + decode selected anchors ======================
__global__ void k_decode(const float* __restrict__ f3, const float* __restrict__ f2,
                         const float* __restrict__ f1, const float* __restrict__ anchors,
                         const unsigned* __restrict__ topidx,
                         float* __restrict__ out, float* __restrict__ offbox) {
  int t = blockIdx.x * blockDim.x + threadIdx.x;   // over B*K
  if (t >= B_ * K_) return;
  int b = t / K_;
  unsigned a = topidx[t];
  int hw;
  const float* ch = chan_base(f3, f2, f1, b, a, &hw);
  size_t s = (size_t)hw;
  float d0 = ch[0], d1 = ch[s], d2 = ch[2 * s], d3 = ch[3 * s];
  float d4 = ch[4 * s], d5 = ch[5 * s], d6 = ch[6 * s];
  float cx = anchors[a], cy = anchors[A_ + a];
  float aw = anchors[2 * (size_t)A_ + a], ah = anchors[3 * (size_t)A_ + a];
  float x = cx + d0 * aw, y = cy + d1 * ah;
  float w = aw * expf(d2), h = ah * expf(d3);
  float obj = fminf(fmaxf(d4, 0.0f), 1.0f);
  float cls = (d6 > d5) ? 1.0f : 0.0f;      // argmax over 2 classes (tie -> 0)
  float x1 = x - 0.5f * w, y1 = y - 0.5f * h;
  float x2 = x1 + w, y2 = y1 + h;
  float* boxes = out;
  float* objo  = out + (size_t)B_ * K_ * 4;
  float* clso  = out + (size_t)B_ * K_ * 5;
  boxes[(size_t)t * 4 + 0] = x1;
  boxes[(size_t)t * 4 + 1] = y1;
  boxes[(size_t)t * 4 + 2] = x2;
  boxes[(size_t)t * 4 + 3] = y2;
  objo[t] = obj;
  clso[t] = cls;
  float off = cls * 10000.0f;
  offbox[(size_t)t * 4 + 0] = x1 + off;
  offbox[(size_t)t * 4 + 1] = y1 + off;
  offbox[(size_t)t * 4 + 2] = x2 + off;
  offbox[(size_t)t * 4 + 3] = y2 + off;
}

// =============== Kernel 4: NMS ==================================================
// TDM stages the box tile into LDS; WMMA computes area-sum tiles; wave32 ballots
// build the 1024x1024 suppression bit-matrix; one wave does the greedy scan.
__global__ __launch_bounds__(1024) void k_nms(const float* __restrict__ offbox,
                                              const unsigned* __restrict__ validf,
                                              float* __restrict__ keep_out) {
  __shared__ float bx1[K_], by1[K_], bx2[K_], by2[K_], ar[K_];
  // TDM staging buffer overlaps the bit-matrix: obs is fully consumed (copied to
  // SoA arrays) before the matrix phase starts writing M16.
  __shared__ union ShU {
    float obs[K_ * 4];                      // 16 KB staged AoS boxes
    unsigned short M16[K_ * 64];            // 128 KB bit-matrix (1024 rows x 64 halfwords)
  } sh;
  __shared__ unsigned validw[32];
  __shared__ unsigned keepw[32];

  int b = blockIdx.x;
  int t = threadIdx.x, lane = t & 31, wid = t >> 5;

  const float* ob = offbox + (size_t)b * K_ * 4;
  float X1, Y1, X2, Y2;

#ifdef HAVE_TDM
  // ---- Tensor Data Mover: DMA 4096 contiguous f32 (16 KB) global -> LDS ----
  if (t < 32) {                             // wave 0 only issues the TDM op
    unsigned long long ga = (unsigned long long)(const void*)ob;
    unsigned lds_addr = (unsigned)(unsigned long long)(void*)sh.obs;
    v4u g0;
    g0.x = 1u;                                              // count=1 (valid user D#)
    g0.y = lds_addr;                                        // bits[63:32]  lds_addr
    g0.z = (unsigned)ga;                                    // bits[95:64]  global_addr lo
    g0.w = ((unsigned)(ga >> 32) & 0x01FFFFFFu) | (2u << 30); // ga hi + type=2
    v8i g1;
    g1[0] = (int)(2u << 16);          // data_size=2 (4 bytes/elem)
    g1[1] = (int)(4096u << 16);       // tensor_dim0[15:0] in bits[63:48]
    g1[2] = (int)(1u << 16);          // tensor_dim0 hi=0 | tensor_dim1=1
    g1[3] = (int)(4096u << 16);       // tensor_dim1 hi=0 | tile_dim0=4096
    g1[4] = 1;                        // tile_dim1=1, tile_dim2=0
    g1[5] = 4096;                     // tensor_dim0_stride lo
    g1[6] = 0;                        // stride hi | tensor_dim1_stride lo16
    g1[7] = 0;
    v4i z4 = {};
#if __clang_major__ >= 23
    v8i z8 = {};
    __builtin_amdgcn_tensor_load_to_lds(g0, g1, z4, z4, z8, 0);
#else
    __builtin_amdgcn_tensor_load_to_lds(g0, g1, z4, z4, 0);
#endif
    __builtin_amdgcn_s_wait_tensorcnt(0);
  }
  __syncthreads();                          // DMA complete + visible to all waves
  X1 = sh.obs[t * 4 + 0];
  Y1 = sh.obs[t * 4 + 1];
  X2 = sh.obs[t * 4 + 2];
  Y2 = sh.obs[t * 4 + 3];
#else
  X1 = ob[(size_t)t * 4 + 0];
  Y1 = ob[(size_t)t * 4 + 1];
  X2 = ob[(size_t)t * 4 + 2];
  Y2 = ob[(size_t)t * 4 + 3];
#endif

  bx1[t] = X1; by1[t] = Y1; bx2[t] = X2; by2[t] = Y2;
  ar[t] = fmaxf(X2 - X1, 0.0f) * fmaxf(Y2 - Y1, 0.0f);
  unsigned v = validf[(size_t)b * K_ + t];
  unsigned vb = ballot32(v != 0u);
  if (lane == 0) validw[wid] = vb;
  __syncthreads();                          // obs fully consumed; M16 may now be written

  // --- parallel phase: suppression bit-matrix, one 16x16 tile at a time ---
  // sum-of-areas tile via V_WMMA_F32_16X16X4_F32: D = [area_i,1] x [1s; area_j]
  for (int tile = wid; tile < 64 * 64; tile += 32) {
    int ti = tile >> 6, tj = tile & 63;
    int i0 = ti * 16, j0 = tj * 16;
    int n = lane & 15;
    int mbase = (lane >> 4) * 8;
    int j = j0 + n;

    float arRow = ar[i0 + n];          // row area for lanes 0..15 (n==lane there)
    float arCol = ar[j];
    v2f Av; Av.x = (lane < 16) ? arRow : 0.0f; Av.y = (lane < 16) ? 1.0f : 0.0f;
    v2f Bv; Bv.x = (lane < 16) ? 1.0f : arCol; Bv.y = 0.0f;
    v8f C = {};
    v8f D = __builtin_amdgcn_wmma_f32_16x16x4_f32(false, Av, false, Bv,
                                                  (short)0, C, false, false);

    float jx1 = bx1[j], jy1 = by1[j], jx2 = bx2[j], jy2 = by2[j];
#pragma unroll
    for (int vv = 0; vv < 8; ++vv) {
      int i = i0 + mbase + vv;
      float xx1 = fmaxf(bx1[i], jx1);
      float yy1 = fmaxf(by1[i], jy1);
      float xx2 = fminf(bx2[i], jx2);
      float yy2 = fminf(by2[i], jy2);
      float inter = fmaxf(xx2 - xx1, 0.0f) * fmaxf(yy2 - yy1, 0.0f);
      // denom = area_i+area_j-inter+1e-9 > 0 always, so
      // iou > 0.5  <=>  inter > 0.5*denom   (no division needed)
      bool pred = (inter > 0.5f * (D[vv] - inter + 1e-9f)) && (j > i);
      unsigned bal = ballot32(pred);   // [15:0]=row i0+vv, [31:16]=row i0+8+vv, cols j0..j0+15
      if (lane == 0) {
        sh.M16[(i0 + vv) * 64 + tj]     = (unsigned short)(bal & 0xFFFFu);
        sh.M16[(i0 + 8 + vv) * 64 + tj] = (unsigned short)(bal >> 16);
      }
    }
  }
  __syncthreads();

  // --- sequential greedy scan: single wave, keep as 32 bit-words ---
  if (wid == 0) {
    unsigned keep = 0xFFFFFFFFu;
    const unsigned* Mw = (const unsigned*)sh.M16;   // row stride = 32 dwords
    for (int i = 0; i < K_; ++i) {
      int w = i >> 5, bit = i & 31;
      unsigned kwv = (unsigned)__shfl((int)keep, w, 32);
      unsigned alive = (kwv >> bit) & (validw[w] >> bit) & 1u;
      if (alive) keep &= ~Mw[i * 32 + lane];
    }
    keepw[lane] = keep & validw[lane];
  }
  __syncthreads();

  unsigned kw = keepw[t >> 5];
  keep_out[(size_t)b * K_ + t] = (float)((kw >> (t & 31)) & 1u);
}

// ================================ launcher ================================
extern "C" void kernel_launch(void* const* d_in, const int* in_sizes, int n_in,
                              void* d_out, int out_size, void* d_ws, size_t ws_size,
                              hipStream_t stream) {
  (void)in_sizes; (void)n_in; (void)out_size; (void)ws_size;
  const float* f3 = (const float*)d_in[0];
  const float* f2 = (const float*)d_in[1];
  const float* f1 = (const float*)d_in[2];
  const float* anchors = (const float*)d_in[3];
  float* out = (float*)d_out;

  char* ws = (char*)d_ws;
  unsigned* keys   = (unsigned*)ws;                                   // B*A
  unsigned* topidx = (unsigned*)(ws + (size_t)B_ * A_ * 4);           // B*K
  unsigned* validf = (unsigned*)(ws + (size_t)B_ * A_ * 4 + (size_t)B_ * K_ * 4);
  float*    offbox = (float*)(ws + (size_t)B_ * A_ * 4 + (size_t)2 * B_ * K_ * 4); // B*K*4

  int totalBA = B_ * A_;
  k_scores<<<(totalBA + 255) / 256, 256, 0, stream>>>(f3, f2, f1, keys);
  k_select<<<B_, 1024, 0, stream>>>(keys, topidx, validf);
  int totalBK = B_ * K_;
  k_decode<<<(totalBK + 255) / 256, 256, 0, stream>>>(f3, f2, f1, anchors, topidx, out, offbox);
  k_nms<<<B_, 1024, 0, stream>>>(offbox, validf, out + (size_t)B_ * K_ * 6);
}